// ConsistencyLoss_68324339745224
// MI455X (gfx1250) — compile-verified
//
#include <hip/hip_runtime.h>

// ---------------- problem constants ----------------
#define N_FFT   512
#define HOP     128
#define QW      7            // 2Q-1 alpha rows
#define KW      1023         // kernel taps
#define BATCH   4
#define LW      131072       // waveform length
#define TFR     1025         // frames
#define TP      1040         // frames padded to multiple of 16
#define NT_TILES 65          // TP/16
#define ROWS    (BATCH*TP)   // 4160
#define ROW_TILES (ROWS/16)  // 260
#define CONV_WAVES (N_FFT*NT_TILES)         // 33280 partial sums
#define KPAD    528          // zero pad each side of K row
#define KSLEN   2080         // padded K row length (covers idx 0..2078)

typedef __attribute__((ext_vector_type(2))) float v2f;
typedef __attribute__((ext_vector_type(8))) float v8f;

__device__ __forceinline__ int reflect_idx(int p) {
  if (p < 0) p = -p;
  if (p >= LW) p = 2 * (LW - 1) - p;
  return p;
}

// ---------------------------------------------------------------------------
// Kernel 1: K[n,tau] = sum_q exp(i*2pi*n*q/512) * alpha[q,tau]   (tiny)
// Kr/Ki stored as [n][1024] (tau=1023 zero pad).
// ---------------------------------------------------------------------------
__global__ void __launch_bounds__(256)
build_K(const float* __restrict__ ar, const float* __restrict__ ai,
        float* __restrict__ Kr, float* __restrict__ Ki) {
  int idx = blockIdx.x * blockDim.x + threadIdx.x;   // 512*1024
  if (idx >= N_FFT * 1024) return;
  int n = idx >> 10;
  int tau = idx & 1023;
  float kr = 0.f, ki = 0.f;
  if (tau < KW) {
#pragma unroll
    for (int qi = 0; qi < QW; ++qi) {
      int q = qi - 3;
      int m = (n * q) % N_FFT;
      if (m < 0) m += N_FFT;
      float th = (float)m * (6.28318530717958647692f / 512.0f);
      float c = __cosf(th), s = __sinf(th);
      float axr = ar[qi * KW + tau];
      float axi = ai[qi * KW + tau];
      kr += c * axr - s * axi;
      ki += s * axr + c * axi;
    }
  }
  Kr[idx] = kr;
  Ki[idx] = ki;
}

// ---------------------------------------------------------------------------
// Kernel 2: STFT (full 512-bin DFT of windowed reflect-padded frames) via
// V_WMMA_F32_16X16X4_F32, with the 16-row frame tile staged in LDS once per
// block and reused by 8 waves (8 n-tiles).
// Output Hri[n][c][t], c = b*2 + (0:Re,1:Im), t padded to TP with zeros.
// ---------------------------------------------------------------------------
__global__ void __launch_bounds__(256)
stft_wmma(const float* __restrict__ wav, const float* __restrict__ win,
          float* __restrict__ Hri) {
  __shared__ __align__(16) float As[16 * N_FFT];   // [row 0..15][k 0..511] 32KB

  const int lane = threadIdx.x & 31;
  const int wave = threadIdx.x >> 5;
  const int rt = blockIdx.x >> 2;                // row tile (0..259)
  const int ng = blockIdx.x & 3;                 // n-tile group
  const int nt = ng * 8 + wave;                  // n tile (0..31)
  const int row0 = rt * 16;
  const int b = row0 / TP;                       // tiles never straddle b
  const int t0 = row0 % TP;

  // ---- stage windowed frames into LDS (once per block) ----
  for (int idx = threadIdx.x; idx < 16 * N_FFT; idx += 256) {
    int i = idx >> 9;             // row in tile
    int k = idx & 511;
    int t = t0 + i;
    float v = 0.f;
    if (t < TFR)
      v = wav[b * LW + reflect_idx(t * HOP + k - 256)] * win[k];
    As[idx] = v;
  }
  __syncthreads();

  const int n0 = nt * 16;
  const int half = lane >> 4;
  const int lr = lane & 15;
  const int ncol = n0 + lr;                      // B/C/D column

  v8f accR = {0.f,0.f,0.f,0.f,0.f,0.f,0.f,0.f};
  v8f accI = {0.f,0.f,0.f,0.f,0.f,0.f,0.f,0.f};
  const float W0 = 6.28318530717958647692f / 512.0f;

  for (int k0 = 0; k0 < N_FFT; k0 += 4) {
    const int ka = k0 + half * 2;                // K-pair base for this lane
    // A fragment from LDS: row M = lane&15, ka even -> 8B aligned
    const float2 av = *(const float2*)&As[lr * N_FFT + ka];
    v2f a; a.x = av.x; a.y = av.y;
    // B fragments: DFT basis cos / -sin with exact integer phase
    int m0 = (ka * ncol) & 511;
    int m1 = ((ka + 1) * ncol) & 511;
    float th0 = (float)m0 * W0, th1 = (float)m1 * W0;
    v2f bc, bs;
    bc.x = __cosf(th0);  bc.y = __cosf(th1);
    bs.x = -__sinf(th0); bs.y = -__sinf(th1);

    accR = __builtin_amdgcn_wmma_f32_16x16x4_f32(false, a, false, bc,
                                                 (short)0, accR, false, false);
    accI = __builtin_amdgcn_wmma_f32_16x16x4_f32(false, a, false, bs,
                                                 (short)0, accI, false, false);
  }

  // D layout: VGPR i -> M = i + half*8, col = n0 + (lane&15)
#pragma unroll
  for (int i = 0; i < 8; ++i) {
    int M = i + half * 8;
    int tt = t0 + M;
    Hri[(size_t)(ncol * 8 + b * 2 + 0) * TP + tt] = accR[i];
    Hri[(size_t)(ncol * 8 + b * 2 + 1) * TP + tt] = accI[i];
  }
}

// ---------------------------------------------------------------------------
// Kernel 3: per-bin complex cross-correlation as Toeplitz GEMM.
// All 8 waves of a block work on the same bin n (different t-tiles) and share:
//   Ks[2][2080] : Kr/Ki row zero-padded 528 each side  -> unpredicated loads
//   Hs[9][1040] : 8 H columns + one all-zero column    -> dead lanes read col 8
// A[t,s] = K[n, 511+s-t]  (XLA conv = correlation, no flip).
// ---------------------------------------------------------------------------
__global__ void __launch_bounds__(256)
conv_wmma(const float* __restrict__ Kr, const float* __restrict__ Ki,
          const float* __restrict__ Hri, float* __restrict__ partials) {
  __shared__ __align__(16) float Ks[2 * KSLEN];   // 16.6KB
  __shared__ __align__(16) float Hs[9 * TP];      // 37.4KB

  const int lane = threadIdx.x & 31;
  const int wave = threadIdx.x >> 5;
  const int n = blockIdx.x / 9;
  const int grp = blockIdx.x % 9;
  const int tt = grp * 8 + wave;                 // 0..71 (65.. idle)

  const float* __restrict__ Krn = Kr + (size_t)n * 1024;
  const float* __restrict__ Kin = Ki + (size_t)n * 1024;
  const float* __restrict__ Hn  = Hri + (size_t)(n * 8) * TP;

  // ---- stage zero-padded K rows ----
  for (int idx = threadIdx.x; idx < 2 * KSLEN; idx += 256) {
    int sel = idx >= KSLEN;
    int j = sel ? (idx - KSLEN) : idx;
    int tau = j - KPAD;
    float v = 0.f;
    if (tau >= 0 && tau < KW) v = sel ? Kin[tau] : Krn[tau];
    Ks[idx] = v;
  }
  // ---- stage H columns (+ zero column 8) ----
  for (int idx = threadIdx.x; idx < 9 * TP; idx += 256) {
    Hs[idx] = (idx < 8 * TP) ? Hn[idx] : 0.f;
  }
  __syncthreads();

  if (tt < NT_TILES) {
    const int t0 = tt * 16;
    const int half = lane >> 4;
    const int lr = lane & 15;
    const int t = t0 + lr;                       // A row
    const int c = lr;                            // B col
    const int cIdx = (c < 8) ? c : 8;            // dead lanes -> zero column
    const int kbase = KPAD + 511 - t;            // Ks index = kbase + s

    v8f acc1 = {0.f,0.f,0.f,0.f,0.f,0.f,0.f,0.f};
    v8f acc2 = {0.f,0.f,0.f,0.f,0.f,0.f,0.f,0.f};

    for (int s0 = 0; s0 < TP; s0 += 4) {
      const int sa = s0 + half * 2;
      const float2 hv = *(const float2*)&Hs[cIdx * TP + sa];  // sa even ✓
      v2f bf; bf.x = hv.x; bf.y = hv.y;
      const int kk = kbase + sa;
      v2f a1, a2;
      a1.x = Ks[kk];           a1.y = Ks[kk + 1];
      a2.x = Ks[KSLEN + kk];   a2.y = Ks[KSLEN + kk + 1];

      acc1 = __builtin_amdgcn_wmma_f32_16x16x4_f32(false, a1, false, bf,
                                                   (short)0, acc1, false, false);
      acc2 = __builtin_amdgcn_wmma_f32_16x16x4_f32(false, a2, false, bf,
                                                   (short)0, acc2, false, false);
    }

    // Cr = (Kr*Hr)[c=2b] - (Ki*Hi)[c=2b+1]
    // Ci = (Ki*Hr)[c=2b] + (Kr*Hi)[c=2b+1]
    float partial = 0.f;
#pragma unroll
    for (int i = 0; i < 8; ++i) {
      float o1 = acc1[i], o2 = acc2[i];
      float p1 = __shfl_xor(o1, 1, 32);          // partner lane's acc1
      float p2 = __shfl_xor(o2, 1, 32);          // partner lane's acc2
      int M = i + half * 8;
      if ((t0 + M) < TFR && (c < 8) && ((c & 1) == 0)) {
        float Cr = o1 - p2;
        float Ci = o2 + p1;
        partial += Cr * Cr + Ci * Ci;
      }
    }
#pragma unroll
    for (int off = 16; off > 0; off >>= 1)
      partial += __shfl_xor(partial, off, 32);
    if (lane == 0) partials[n * NT_TILES + tt] = partial;
  }
}

// ---------------------------------------------------------------------------
// Kernel 4: deterministic final reduction (single block, fixed order, f64).
// ---------------------------------------------------------------------------
__global__ void __launch_bounds__(256)
reduce_loss(const float* __restrict__ partials, float* __restrict__ out) {
  __shared__ double sm[256];
  double acc = 0.0;
  for (int i = threadIdx.x; i < CONV_WAVES; i += 256)
    acc += (double)partials[i];
  sm[threadIdx.x] = acc;
  __syncthreads();
  for (int s = 128; s > 0; s >>= 1) {
    if ((int)threadIdx.x < s) sm[threadIdx.x] += sm[threadIdx.x + s];
    __syncthreads();
  }
  if (threadIdx.x == 0)
    out[0] = (float)(sm[0] / (double)(BATCH * TFR));
}

// ---------------------------------------------------------------------------
extern "C" void kernel_launch(void* const* d_in, const int* in_sizes, int n_in,
                              void* d_out, int out_size, void* d_ws, size_t ws_size,
                              hipStream_t stream) {
  (void)in_sizes; (void)n_in; (void)out_size; (void)ws_size;
  const float* wav = (const float*)d_in[0];   // (4, 131072) f32
  const float* win = (const float*)d_in[1];   // (512,) f32
  const float* ar  = (const float*)d_in[2];   // (7, 1023) f32
  const float* ai  = (const float*)d_in[3];   // (7, 1023) f32
  float* out = (float*)d_out;

  float* Kr       = (float*)d_ws;                      // 512*1024
  float* Ki       = Kr + (size_t)N_FFT * 1024;         // 512*1024
  float* Hri      = Ki + (size_t)N_FFT * 1024;         // 512*8*TP
  float* partials = Hri + (size_t)N_FFT * 8 * TP;      // CONV_WAVES

  build_K<<<(N_FFT * 1024 + 255) / 256, 256, 0, stream>>>(ar, ai, Kr, Ki);
  stft_wmma<<<ROW_TILES * 4, 256, 0, stream>>>(wav, win, Hri);
  conv_wmma<<<N_FFT * 9, 256, 0, stream>>>(Kr, Ki, Hri, partials);
  reduce_loss<<<1, 256, 0, stream>>>(partials, out);
}